// AttentionMessagePassing_40149354283101
// MI455X (gfx1250) — compile-verified
//
#include <hip/hip_runtime.h>
#include <hip/hip_bf16.h>

#define DIM 128
#define LDA 132   // padded LDS row stride (floats) to avoid 16-way bank conflicts

typedef __attribute__((ext_vector_type(2))) float v2f;
typedef __attribute__((ext_vector_type(8))) float v8f;

// ---------------------------------------------------------------------------
// GEMM: out[r, :] = A[r, :] @ W + bias (+ residual[r, :] if HasResidual)
// A: [nrows, 128], W: [128, 128] row-major, bias: [128]
// Block = 256 threads = 8 waves. Block tile: 64 rows x 128 cols.
// Wave w computes cols [16w, 16w+16) for four 16-row tiles, using
// v_wmma_f32_16x16x4_f32 over K=128 in 32 steps (B fragment reused 4x).
// Full tiles take an unguarded epilogue; only the tail block bounds-checks.
// ---------------------------------------------------------------------------
template <bool HasResidual>
__global__ __launch_bounds__(256)
void gat_gemm128(const float* __restrict__ A, const float* __restrict__ W,
                 const float* __restrict__ bias, const float* __restrict__ residual,
                 float* __restrict__ out, int nrows)
{
    __shared__ float sA[64 * LDA];

    const int tid  = threadIdx.x;
    const int lane = tid & 31;
    const int wave = tid >> 5;
    const int half = lane >> 4;      // 0: lanes 0-15, 1: lanes 16-31
    const int l15  = lane & 15;
    const int row0 = blockIdx.x * 64;
    const bool full = (row0 + 64 <= nrows);

    // Stage A tile (64 x 128 fp32) into LDS: 2048 float4, 8 per thread.
    if (full) {
        #pragma unroll
        for (int i = 0; i < 8; ++i) {
            const int linear = tid + 256 * i;       // float4 index within tile
            const int r  = linear >> 5;             // 32 float4 per row
            const int c4 = linear & 31;
            float4 val = *((const float4*)(A + (size_t)(row0 + r) * DIM) + c4);
            *(float4*)&sA[r * LDA + c4 * 4] = val;
        }
    } else {
        #pragma unroll
        for (int i = 0; i < 8; ++i) {
            const int linear = tid + 256 * i;
            const int r  = linear >> 5;
            const int c4 = linear & 31;
            float4 val = make_float4(0.f, 0.f, 0.f, 0.f);
            if (row0 + r < nrows) {
                val = *((const float4*)(A + (size_t)(row0 + r) * DIM) + c4);
            }
            *(float4*)&sA[r * LDA + c4 * 4] = val;
        }
    }
    __syncthreads();

    const int col0 = wave * 16;
    const int col  = col0 + l15;

    v8f c0 = {}, c1 = {}, c2 = {}, c3 = {};

    #pragma unroll 4
    for (int kk = 0; kk < 32; ++kk) {
        const int ko = kk * 4 + half * 2;
        // B fragment: b.x = W[ko][col], b.y = W[ko+1][col]   (4x16 tile)
        v2f b;
        b.x = W[(size_t)ko * DIM + col];
        b.y = W[(size_t)(ko + 1) * DIM + col];
        // A fragments: a.x = A[m*16 + l15][ko], a.y = A[...][ko+1]  (16x4 tiles)
        v2f a0, a1, a2, a3;
        a0.x = sA[(0 * 16 + l15) * LDA + ko];  a0.y = sA[(0 * 16 + l15) * LDA + ko + 1];
        a1.x = sA[(1 * 16 + l15) * LDA + ko];  a1.y = sA[(1 * 16 + l15) * LDA + ko + 1];
        a2.x = sA[(2 * 16 + l15) * LDA + ko];  a2.y = sA[(2 * 16 + l15) * LDA + ko + 1];
        a3.x = sA[(3 * 16 + l15) * LDA + ko];  a3.y = sA[(3 * 16 + l15) * LDA + ko + 1];

        c0 = __builtin_amdgcn_wmma_f32_16x16x4_f32(false, a0, false, b, (short)0, c0, false, false);
        c1 = __builtin_amdgcn_wmma_f32_16x16x4_f32(false, a1, false, b, (short)0, c1, false, false);
        c2 = __builtin_amdgcn_wmma_f32_16x16x4_f32(false, a2, false, b, (short)0, c2, false, false);
        c3 = __builtin_amdgcn_wmma_f32_16x16x4_f32(false, a3, false, b, (short)0, c3, false, false);
    }

    const float bcol = bias[col];
    v8f acc[4] = { c0, c1, c2, c3 };

    if (full) {
        // Unguarded epilogue: no per-element compare/branch; loads/stores clause.
        #pragma unroll
        for (int m = 0; m < 4; ++m) {
            #pragma unroll
            for (int r = 0; r < 8; ++r) {
                // C/D layout: VGPR r -> row (r + 8*half), col = lane&15
                const int row = row0 + m * 16 + r + half * 8;
                const size_t idx = (size_t)row * DIM + col;
                float val = acc[m][r] + bcol;
                if constexpr (HasResidual) val += residual[idx];
                out[idx] = val;
            }
        }
    } else {
        #pragma unroll
        for (int m = 0; m < 4; ++m) {
            #pragma unroll
            for (int r = 0; r < 8; ++r) {
                const int row = row0 + m * 16 + r + half * 8;
                if (row < nrows) {
                    const size_t idx = (size_t)row * DIM + col;
                    float val = acc[m][r] + bcol;
                    if constexpr (HasResidual) val += residual[idx];
                    out[idx] = val;
                }
            }
        }
    }
}

// ---------------------------------------------------------------------------
// Zero-fill kernel (aggregated buffer), float4 granularity.
// ---------------------------------------------------------------------------
__global__ __launch_bounds__(256)
void gat_zero(float4* __restrict__ p, long long n4)
{
    long long i = (long long)blockIdx.x * blockDim.x + threadIdx.x;
    if (i < n4) p[i] = make_float4(0.f, 0.f, 0.f, 0.f);
}

// ---------------------------------------------------------------------------
// Edge kernel: one wave32 per edge.
// Lane l holds dims [4l, 4l+4) of the 128-dim vectors; head h = l/8
// (32 dims = 8 lanes per head, 4 heads per wave -- exact wave32 fit).
//   score[h] = dot(q[s],k[r]) per head * 1/sqrt(32)
//   attn     = softmax over the 4 heads (xor-shuffle reduce over lanes 8,16)
//   atomicAdd aggregated[r] += v[s] * attn[h]
// ---------------------------------------------------------------------------
__global__ __launch_bounds__(256)
void gat_edge(const float* __restrict__ q, const float* __restrict__ k,
              const float* __restrict__ v, const int* __restrict__ senders,
              const int* __restrict__ receivers, float* __restrict__ agg,
              int n_edges)
{
    const int lane = threadIdx.x & 31;
    const int wave = threadIdx.x >> 5;
    const int e = blockIdx.x * 8 + wave;
    if (e >= n_edges) return;

    const int s = senders[e];
    const int r = receivers[e];

    const float4 qv = *((const float4*)(q + (size_t)s * DIM) + lane);
    const float4 kv = *((const float4*)(k + (size_t)r * DIM) + lane);

    // partial dot over this lane's 4 dims
    float p = qv.x * kv.x + qv.y * kv.y + qv.z * kv.z + qv.w * kv.w;
    // reduce within the 8-lane head group
    p += __shfl_xor(p, 1);
    p += __shfl_xor(p, 2);
    p += __shfl_xor(p, 4);
    p *= 0.17677669529663687f;   // 1/sqrt(HEAD_DIM=32)

    // softmax over the 4 heads: orbit {l, l^8, l^16, l^24} touches each head once
    float m = p;
    m = fmaxf(m, __shfl_xor(m, 8));
    m = fmaxf(m, __shfl_xor(m, 16));
    float ex = __expf(p - m);
    float sum = ex;
    sum += __shfl_xor(sum, 8);
    sum += __shfl_xor(sum, 16);
    const float attn = ex / sum;

    const float4 vv = *((const float4*)(v + (size_t)s * DIM) + lane);
    float* dst = agg + (size_t)r * DIM + lane * 4;
    atomicAdd(dst + 0, vv.x * attn);
    atomicAdd(dst + 1, vv.y * attn);
    atomicAdd(dst + 2, vv.z * attn);
    atomicAdd(dst + 3, vv.w * attn);
}

// ---------------------------------------------------------------------------
extern "C" void kernel_launch(void* const* d_in, const int* in_sizes, int n_in,
                              void* d_out, int out_size, void* d_ws, size_t ws_size,
                              hipStream_t stream)
{
    const float* nodes     = (const float*)d_in[0];
    const int*   senders   = (const int*)  d_in[1];
    const int*   receivers = (const int*)  d_in[2];
    const float* Wq = (const float*)d_in[3];  const float* bq = (const float*)d_in[4];
    const float* Wk = (const float*)d_in[5];  const float* bk = (const float*)d_in[6];
    const float* Wv = (const float*)d_in[7];  const float* bv = (const float*)d_in[8];
    const float* Wo = (const float*)d_in[9];  const float* bo = (const float*)d_in[10];
    float* out = (float*)d_out;

    const int n_nodes = in_sizes[0] / DIM;
    const int n_edges = in_sizes[1];

    const size_t per = (size_t)n_nodes * DIM;
    float* q   = (float*)d_ws;
    float* kk  = q  + per;
    float* v   = kk + per;
    float* agg = v  + per;

    const int mblocks = (n_nodes + 63) / 64;

    // Q/K/V projections
    gat_gemm128<false><<<mblocks, 256, 0, stream>>>(nodes, Wq, bq, nullptr, q,  n_nodes);
    gat_gemm128<false><<<mblocks, 256, 0, stream>>>(nodes, Wk, bk, nullptr, kk, n_nodes);
    gat_gemm128<false><<<mblocks, 256, 0, stream>>>(nodes, Wv, bv, nullptr, v,  n_nodes);

    // zero the segment-sum accumulator
    const long long n4 = (long long)per / 4;
    gat_zero<<<(unsigned)((n4 + 255) / 256), 256, 0, stream>>>((float4*)agg, n4);

    // per-edge attention + scatter-add
    gat_edge<<<(n_edges + 7) / 8, 256, 0, stream>>>(q, kk, v, senders, receivers, agg, n_edges);

    // output projection + bias + residual
    gat_gemm128<true><<<mblocks, 256, 0, stream>>>(agg, Wo, bo, nodes, out, n_nodes);
}